// G1_sub2_and_sub3_update_84937273245886
// MI455X (gfx1250) — compile-verified
//
#include <hip/hip_runtime.h>

// CDNA5 / gfx1250, wave32.
typedef __attribute__((ext_vector_type(2))) float v2f;
typedef __attribute__((ext_vector_type(8))) float v8f;

// ---------------------------------------------------------------------------
// sub2: per type column c, msg2[c] = sum over its deg2 edges of
//       emb[left_specific[row2[e]]]; out[right_common[c]] =
//       emb[right_common[c]] + msg2[c] + (n_ent - deg_c).
// One wave per (type, 16-column tile). The 64-row reduction uses
// V_WMMA_F32_16X16X4_F32 with an all-ones A matrix: D = 1 x B + C sums the
// 4 gathered rows (B, 4x16 fp32) into the fp32 accumulator C, chained over
// deg2/4 chunks. Exact fp32; A==1 makes the K/M lane permutation irrelevant.
// Loads are unconditional (masked with selects afterwards) so no EXEC
// divergence exists anywhere in the WMMA loop and gathers can pipeline.
// ---------------------------------------------------------------------------
__global__ void sub2_wmma_kernel(const float* __restrict__ emb,
                                 const int* __restrict__ row2,
                                 const int* __restrict__ col2,
                                 const int* __restrict__ left_specific,
                                 const int* __restrict__ right_common,
                                 float* __restrict__ out,
                                 int n_ent, int D, int deg2)
{
    const int c    = blockIdx.x;          // type column
    const int tile = threadIdx.x >> 5;    // which 16-column tile of D
    const int lane = threadIdx.x & 31;
    const int n    = lane & 15;           // column within tile
    const int half = lane >> 4;           // selects K slots {0,1} vs {2,3}
    const int coff = tile * 16 + n;

    v8f acc = {};                         // C/D accumulator, starts at 0
    v2f a;  a.x = 1.0f; a.y = 1.0f;       // ones A-matrix (16x4, all lanes 1)

    int degc = 0;
    float rem = 0.0f;                     // scalar tail if deg2 % 4 != 0
    const long long ebase = (long long)c * deg2;
    const int nchunk = deg2 >> 2;

#pragma unroll 4
    for (int kk = 0; kk < nchunk; ++kk) {
        const long long e = ebase + ((long long)kk << 2);
        // wave-uniform edge metadata (scalar loads)
        const int c0 = col2[e + 0], c1 = col2[e + 1];
        const int c2 = col2[e + 2], c3 = col2[e + 3];
        const int r0 = row2[e + 0], r1 = row2[e + 1];
        const int r2 = row2[e + 2], r3 = row2[e + 3];
        // wave-uniform source node ids
        const int s0 = left_specific[r0];
        const int s1 = left_specific[r1];
        const int s2 = left_specific[r2];
        const int s3 = left_specific[r3];
        degc += (c0 == c) + (c1 == c) + (c2 == c) + (c3 == c);

        // each half-wave supplies two of the four K rows of B (4x16 fp32)
        const int sa = half ? s2 : s0;
        const int sb = half ? s3 : s1;
        const bool ma = (half ? c2 : c0) == c;
        const bool mb = (half ? c3 : c1) == c;
        // unconditional gathers; mask with selects (no EXEC manipulation)
        const float va = emb[(size_t)sa * D + coff];
        const float vb = emb[(size_t)sb * D + coff];
        v2f b;
        b.x = ma ? va : 0.0f;
        b.y = mb ? vb : 0.0f;

        // v_wmma_f32_16x16x4_f32: acc = ones * B + acc
        acc = __builtin_amdgcn_wmma_f32_16x16x4_f32(
            /*neg_a=*/false, a, /*neg_b=*/false, b,
            /*c_mod=*/(short)0, acc, /*reuse_a=*/false, /*reuse_b=*/false);
    }
    // tail edges (deg2 not a multiple of 4) -- not hit for deg2 == 64
    for (int t = nchunk << 2; t < deg2; ++t) {
        const long long e = ebase + t;
        const int cc = col2[e];
        const float v = emb[(size_t)left_specific[row2[e]] * D + coff];
        if (cc == c) { ++degc; rem += v; }
    }

    // D-matrix row M=0: element 0 of the v8f on lanes 0..15 (N = lane)
    if (lane < 16) {
        const int dst = right_common[c];
        const size_t idx = (size_t)dst * D + coff;
        out[idx] = emb[idx] + acc[0] + rem + ((float)n_ent - (float)degc);
    }
}

// ---------------------------------------------------------------------------
// sub3: per entity column e, gather its deg3 type rows from the *updated*
// table (already written into `out` by sub2), then scale the entity row:
//   out[rs] = emb[rs] * (1 - (sum + (n_typ - deg)) / (1 + deg))
// One wave per entity row; each lane owns 4 consecutive floats -> b128
// loads/stores, the bandwidth-optimal path for the dominant ~200 MB stream.
// Gathers hit the 512 KB type table (L2-resident). Assumes D == 128.
// ---------------------------------------------------------------------------
__global__ void sub3_kernel(const float* __restrict__ emb,
                            const int* __restrict__ row3,
                            const int* __restrict__ col3,
                            const int* __restrict__ left_common,
                            const int* __restrict__ right_specific,
                            float* __restrict__ out,
                            int n_ent, int n_typ, int D, int deg3)
{
    const int wave = blockIdx.x * (blockDim.x >> 5) + (threadIdx.x >> 5);
    const int lane = threadIdx.x & 31;
    if (wave >= n_ent) return;
    const int e    = wave;
    const int coff = lane << 2;           // lane*4, D == 128

    float4 s = make_float4(0.f, 0.f, 0.f, 0.f);
    int degc = 0;
    const long long ebase = (long long)e * deg3;
#pragma unroll 4
    for (int k = 0; k < deg3; ++k) {
        const int cc = col3[ebase + k];
        const int rr = row3[ebase + k];
        const int g  = left_common[rr];            // updated (type) node id
        const float m = (cc == e) ? 1.0f : 0.0f;
        degc += (cc == e);
        const float4 v = *(const float4*)(out + (size_t)g * D + coff);
        s.x += v.x * m; s.y += v.y * m; s.z += v.z * m; s.w += v.w * m;
    }

    const float addc = (float)n_typ - (float)degc;
    const float inv  = 1.0f / (1.0f + (float)degc);
    const int rs = right_specific[e];
    const size_t idx = (size_t)rs * D + coff;
    const float4 xin = *(const float4*)(emb + idx);
    float4 r;
    r.x = xin.x * (1.0f - (s.x + addc) * inv);
    r.y = xin.y * (1.0f - (s.y + addc) * inv);
    r.z = xin.z * (1.0f - (s.z + addc) * inv);
    r.w = xin.w * (1.0f - (s.w + addc) * inv);
    *(float4*)(out + idx) = r;
}

extern "C" void kernel_launch(void* const* d_in, const int* in_sizes, int n_in,
                              void* d_out, int out_size, void* d_ws, size_t ws_size,
                              hipStream_t stream) {
    const float* emb            = (const float*)d_in[0];
    const int*   row2           = (const int*)d_in[1];
    const int*   col2           = (const int*)d_in[2];
    const int*   row3           = (const int*)d_in[3];
    const int*   col3           = (const int*)d_in[4];
    const int*   left_specific  = (const int*)d_in[5];
    const int*   right_common   = (const int*)d_in[6];
    const int*   left_common    = (const int*)d_in[7];
    const int*   right_specific = (const int*)d_in[8];
    float* out = (float*)d_out;

    const int n_ent   = in_sizes[5];                 // 200000
    const int n_typ   = in_sizes[7];                 // 1000
    const int n_nodes = n_ent + n_typ;
    const int D       = out_size / n_nodes;          // 128
    const int deg2    = in_sizes[1] / n_typ;         // 64
    const int deg3    = in_sizes[3] / n_ent;         // 4

    // sub2: one block per type; (D/16) waves, one per 16-column tile.
    const int ntiles = D / 16;                       // 8 -> 256 threads
    sub2_wmma_kernel<<<n_typ, ntiles * 32, 0, stream>>>(
        emb, row2, col2, left_specific, right_common, out, n_ent, D, deg2);

    // sub3: one wave per entity row, 8 waves (256 threads) per block.
    const int wpb = 8;
    const int blocks = (n_ent + wpb - 1) / wpb;
    sub3_kernel<<<blocks, wpb * 32, 0, stream>>>(
        emb, row3, col3, left_common, right_specific, out, n_ent, n_typ, D, deg3);
}